// GCN_29265907155119
// MI455X (gfx1250) — compile-verified
//
#include <hip/hip_runtime.h>
#include <hip/hip_bf16.h>

// 2-layer GCN (GCNConv x2 + log_softmax) for MI455X / gfx1250.
//
// Pipeline (all on `stream`):
//  1) deg init(=1 self loop) + edge atomic deg accum -> dinv = rsqrt(deg)
//  2) h1 = x @ W1 via V_WMMA_F32_16X16X4_F32 (one 16x16 f32 tile per wave32)
//  3) agg1[i] = dinv[i]^2 * h1[i]        (self-loop term, no atomics)
//  4) edges: agg1[dst] += dinv[s]*dinv[d]*h1[src]  (16 lanes per edge, 64B gather)
//  5) per node: h2 = sum_f relu(agg1+b1)*W2[f];  agg2[i] = dinv^2*h2 (fused)
//  6) edges: agg2[dst] += norm*h2[src]   (1 atomic per edge)
//  7) out = log_softmax over F_OUT=1 logit == v - v (identically 0 for finite v)
//
// Roofline: GEMM is 0.4 GFLOP and bound by the 51.2 MB read of x (~2.2us at
// 23.3 TB/s). The edge phase dominates: h1/agg1 are 6.4 MB each -> fully
// L2-resident (192 MB), so the 3.2M-edge gather (64 B/edge) and 51.2M f32
// atomics run at L2 speed, with only x + edge_index (~77 MB) touching HBM.

typedef __attribute__((ext_vector_type(2))) float v2f;
typedef __attribute__((ext_vector_type(8))) float v8f;

#define F_IN  128
#define F_HID 16

// ---------------- degree / normalization ----------------

__global__ void gcn_deg_init(float* __restrict__ deg, int n) {
    int i = blockIdx.x * blockDim.x + threadIdx.x;
    if (i < n) deg[i] = 1.0f;   // self loop contributes 1
}

__global__ void gcn_deg_accum(const int* __restrict__ dst, float* __restrict__ deg, int e) {
    int i = blockIdx.x * blockDim.x + threadIdx.x;
    if (i < e) atomicAdd(&deg[dst[i]], 1.0f);
}

__global__ void gcn_dinv(const float* __restrict__ deg, float* __restrict__ dinv, int n) {
    int i = blockIdx.x * blockDim.x + threadIdx.x;
    if (i < n) {
        float d = deg[i];
        dinv[i] = (d > 0.0f) ? rsqrtf(fmaxf(d, 1.0f)) : 0.0f;
    }
}

// ---------------- layer-1 GEMM: h1 = x @ W1 (f32 WMMA) ----------------
// One wave computes one 16-row tile of h1 (16x16 output, K=128 in steps of 4).
// A (16x4 f32): lane L -> M = L%16, K pair = k + 2*(L/16).
// B (4x16 f32): lane L -> N = L%16, K pair = k + 2*(L/16).
// D (16x16 f32): VGPR r -> M = r + 8*(L/16), N = L%16.

__global__ __launch_bounds__(32) void gcn_gemm1_wmma(
    const float* __restrict__ x, const float* __restrict__ W1,
    float* __restrict__ h1, int n)
{
    const int lane = threadIdx.x;         // 0..31, wave32
    const int m    = lane & 15;           // row within tile / column index
    const int hi   = lane >> 4;           // 0 or 1 -> K-pair select
    const int row0 = blockIdx.x * 16;

    int row = row0 + m;
    if (row > n - 1) row = n - 1;         // defensive clamp (n is a multiple of 16)
    const float* __restrict__ xrow = x + (size_t)row * F_IN;

    v8f acc = {};
    #pragma unroll 4
    for (int k = 0; k < F_IN; k += 4) {
        const int ka = k + 2 * hi;
        v2f a, b;
        a.x = xrow[ka + 0];
        a.y = xrow[ka + 1];
        b.x = W1[(ka + 0) * F_HID + m];   // W1 row-major [128][16]
        b.y = W1[(ka + 1) * F_HID + m];
        acc = __builtin_amdgcn_wmma_f32_16x16x4_f32(
            /*neg_a=*/false, a, /*neg_b=*/false, b,
            /*c_mod=*/(short)0, acc, /*reuse_a=*/false, /*reuse_b=*/false);
    }

    // Store D. Fast path: full tile (always taken for n % 16 == 0) -> single
    // uniform scalar branch, no per-row EXEC manipulation. Each store writes
    // two contiguous 64B row segments (lanes 0-15: row r, lanes 16-31: row r+8).
    float* __restrict__ out = h1 + (size_t)row0 * F_HID;
    if (row0 + 16 <= n) {
        #pragma unroll
        for (int r = 0; r < 8; ++r) {
            out[(r + 8 * hi) * F_HID + m] = acc[r];
        }
    } else {
        #pragma unroll
        for (int r = 0; r < 8; ++r) {
            const int M = r + 8 * hi;
            if (row0 + M < n) out[M * F_HID + m] = acc[r];
        }
    }
}

// ---------------- aggregation, layer 1 ----------------

__global__ void gcn_agg1_init(const float* __restrict__ h1, const float* __restrict__ dinv,
                              float* __restrict__ agg1, int n) {
    int t = blockIdx.x * blockDim.x + threadIdx.x;   // n * 16 elements
    if (t >= n * F_HID) return;
    int i = t >> 4;
    float di = dinv[i];
    agg1[t] = di * di * h1[t];                       // self-loop norm = dinv^2
}

__global__ void gcn_edge_agg1(const int* __restrict__ src, const int* __restrict__ dst,
                              const float* __restrict__ dinv, const float* __restrict__ h1,
                              float* __restrict__ agg1, int e)
{
    int t = blockIdx.x * blockDim.x + threadIdx.x;   // 16 lanes per edge
    int edge = t >> 4;
    if (edge >= e) return;
    int f = t & 15;
    int s = src[edge];
    int d = dst[edge];
    float norm = dinv[s] * dinv[d];                  // broadcast within half-wave
    atomicAdd(&agg1[d * F_HID + f], norm * h1[s * F_HID + f]);
}

// ---------------- layer 2: bias + relu + 16->1 linear, fused ----------------

__global__ void gcn_node2(const float* __restrict__ agg1, const float* __restrict__ b1,
                          const float* __restrict__ W2, const float* __restrict__ dinv,
                          float* __restrict__ h2, float* __restrict__ agg2, int n)
{
    int i = blockIdx.x * blockDim.x + threadIdx.x;
    if (i >= n) return;
    const float* a = agg1 + (size_t)i * F_HID;
    float s = 0.0f;
    #pragma unroll
    for (int f = 0; f < F_HID; ++f)
        s += fmaxf(a[f] + b1[f], 0.0f) * W2[f];
    h2[i] = s;
    float di = dinv[i];
    agg2[i] = di * di * s;                           // self-loop term of layer 2
}

__global__ void gcn_edge_agg2(const int* __restrict__ src, const int* __restrict__ dst,
                              const float* __restrict__ dinv, const float* __restrict__ h2,
                              float* __restrict__ agg2, int e)
{
    int i = blockIdx.x * blockDim.x + threadIdx.x;
    if (i >= e) return;
    int s = src[i];
    int d = dst[i];
    atomicAdd(&agg2[d], dinv[s] * dinv[d] * h2[s]);
}

__global__ void gcn_final(const float* __restrict__ agg2, const float* __restrict__ b2,
                          float* __restrict__ out, int n)
{
    int i = blockIdx.x * blockDim.x + threadIdx.x;
    if (i >= n) return;
    // log_softmax over a single logit: v - logsumexp(v) == v - v (exactly 0 for finite v)
    float v = agg2[i] + b2[0];
    out[i] = v - v;
}

// ---------------- launcher ----------------

extern "C" void kernel_launch(void* const* d_in, const int* in_sizes, int n_in,
                              void* d_out, int out_size, void* d_ws, size_t ws_size,
                              hipStream_t stream)
{
    const float* x   = (const float*)d_in[0];
    const int*   ei  = (const int*)d_in[1];
    const float* W1  = (const float*)d_in[2];
    const float* b1  = (const float*)d_in[3];
    const float* W2  = (const float*)d_in[4];
    const float* b2  = (const float*)d_in[5];
    float* out = (float*)d_out;

    const int n = in_sizes[0] / F_IN;       // 100000
    const int e = in_sizes[1] / 2;          // 3200000
    const int* src = ei;
    const int* dst = ei + e;

    // workspace carve-up (floats)
    float* deg  = (float*)d_ws;             // n
    float* dinv = deg  + n;                 // n
    float* h1   = dinv + n;                 // n*16
    float* agg1 = h1   + (size_t)n * F_HID; // n*16
    float* h2   = agg1 + (size_t)n * F_HID; // n
    float* agg2 = h2   + n;                 // n

    const int B = 256;
    const int gN  = (n + B - 1) / B;
    const int gE  = (e + B - 1) / B;
    const int gNF = (n * F_HID + B - 1) / B;
    const int gEF = (int)(((long long)e * F_HID + B - 1) / B);

    gcn_deg_init <<<gN, B, 0, stream>>>(deg, n);
    gcn_deg_accum<<<gE, B, 0, stream>>>(dst, deg, e);
    gcn_dinv     <<<gN, B, 0, stream>>>(deg, dinv, n);

    gcn_gemm1_wmma<<<(n + 15) / 16, 32, 0, stream>>>(x, W1, h1, n);

    gcn_agg1_init<<<gNF, B, 0, stream>>>(h1, dinv, agg1, n);
    gcn_edge_agg1<<<gEF, B, 0, stream>>>(src, dst, dinv, h1, agg1, e);

    gcn_node2    <<<gN, B, 0, stream>>>(agg1, b1, W2, dinv, h2, agg2, n);
    gcn_edge_agg2<<<gE, B, 0, stream>>>(src, dst, dinv, h2, agg2, e);

    gcn_final    <<<gN, B, 0, stream>>>(agg2, b2, out, n);
}